// EMDLoss_36885179138307
// MI455X (gfx1250) — compile-verified
//
#include <hip/hip_runtime.h>
#include <math.h>

// ---------------------------------------------------------------------------
// Sinkhorn EMD, B=16, N=M=2048, D=3, eps=0.005, 100 iters, tol=1e-3.
// Strategy (MI455X / gfx1250):
//  * never materialize the 256MB dist matrix (it exceeds the 192MB L2);
//    recompute 16x16 distance tiles with V_WMMA_F32_16X16X4_F32, folding
//    |x|^2+|y|^2 into the WMMA C operand and -2 into the A operand so the
//    WMMA output IS the squared-distance tile.
//  * stage the full column-side data (points, |y|^2, potential) for one
//    batch in LDS (40KB of the 320KB/WGP) -- inner loop touches only LDS.
//  * numerically stable online logsumexp kept in base-2 so v_exp_f32 /
//    v_log_f32 (natively 2^x / log2) are used directly; 1 fma folds the
//    1/eps*log2e scale.
//  * WMMA C-tile layout: row r on lanes 0-15 (VGPR r), row 8+r on lanes
//    16-31 => per-row reduction is 4 shfl_xor steps (masks 1,2,4,8).
//  * convergence freeze reproduced with a device flag + atomicMax on float
//    bit patterns (order-independent => deterministic); final sum is a
//    fixed-order serial reduction (bitwise deterministic across replays).
// ---------------------------------------------------------------------------

namespace {
constexpr int   kB            = 16;
constexpr int   kN            = 2048;
constexpr int   kM            = 2048;
constexpr float kEps          = 0.005f;
constexpr int   kMaxIter      = 100;
constexpr float kTol          = 1e-3f;
constexpr float kLog2e        = 1.44269504088896340736f;
constexpr float kLn2          = 0.69314718055994530942f;
constexpr float kK            = kLog2e / kEps;   // (1/eps) in base-2 units

constexpr int kRowsPerWave    = 16;
constexpr int kWavesPerBlock  = 8;
constexpr int kThreads        = kWavesPerBlock * 32;                 // 256
constexpr int kRowsPerBlock   = kRowsPerWave * kWavesPerBlock;       // 128
constexpr int kBlocksPerBatch = kN / kRowsPerBlock;                  // 16
static_assert(kN == kM, "shared-LDS column staging assumes N == M");
}  // namespace

typedef float v2f __attribute__((ext_vector_type(2)));
typedef float v8f __attribute__((ext_vector_type(8)));

// -------------------------- init: sq-norms, zero state ---------------------
__global__ void emd_init(const float* __restrict__ p1,
                         const float* __restrict__ p2,
                         float* __restrict__ u, float* __restrict__ v,
                         float* __restrict__ sq1, float* __restrict__ sq2,
                         float* __restrict__ partials,
                         unsigned* __restrict__ flags) {
  int i = blockIdx.x * blockDim.x + threadIdx.x;
  if (i < kB * kN) {
    float a0 = p1[i * 3 + 0], a1 = p1[i * 3 + 1], a2 = p1[i * 3 + 2];
    sq1[i] = a0 * a0 + a1 * a1 + a2 * a2;
    u[i] = 0.0f;
    float b0 = p2[i * 3 + 0], b1 = p2[i * 3 + 1], b2 = p2[i * 3 + 2];
    sq2[i] = b0 * b0 + b1 * b1 + b2 * b2;
    v[i] = 0.0f;
  }
  if (i < kB * kBlocksPerBatch) partials[i] = 0.0f;
  if (i < 4) flags[i] = 0u;
}

// ---------------- one Sinkhorn half-update (rows <- logsumexp over cols) ---
// x_new[row] = eps * (logw - logsumexp_col( (y[col] - dist(row,col)) / eps ))
__global__ __launch_bounds__(kThreads) void sinkhorn_update(
    const float* __restrict__ pa,   // (B, R, 3) row-side points
    const float* __restrict__ pb,   // (B, C, 3) col-side points
    const float* __restrict__ sqa,  // (B, R)
    const float* __restrict__ sqb,  // (B, C)
    const float* __restrict__ y,    // (B, C) other potential
    float* __restrict__ x,          // (B, R) potential being updated
    unsigned* __restrict__ maxdiff, // bit-encoded max |x_new - x_old|
    const unsigned* __restrict__ done,
    float logw) {
  if (*done) return;  // frozen after convergence (matches reference)

  __shared__ float sP[kM * 3];   // 24 KB  column points
  __shared__ float sSQ[kM];      //  8 KB  column sq-norms
  __shared__ float sYK[kM];      //  8 KB  y[col] * (log2e/eps)

  const int b    = blockIdx.y;
  const int lane = threadIdx.x & 31;
  const int wave = threadIdx.x >> 5;
  const int row0 = blockIdx.x * kRowsPerBlock + wave * kRowsPerWave;

  const float* PA = pa  + (size_t)b * kN * 3;
  const float* PB = pb  + (size_t)b * kM * 3;
  const float* SA = sqa + (size_t)b * kN;
  const float* SB = sqb + (size_t)b * kM;
  const float* Y  = y   + (size_t)b * kM;
  float*       X  = x   + (size_t)b * kN;

  // cooperative stage of column-side data into LDS (read-only afterwards)
  for (int i = threadIdx.x; i < kM; i += kThreads) {
    sSQ[i] = SB[i];
    sYK[i] = Y[i] * kK;
  }
  for (int i = threadIdx.x; i < kM * 3; i += kThreads) sP[i] = PB[i];
  __syncthreads();

  const bool hi  = lane >= 16;
  const int  l16 = lane & 15;

  // A operand (16x4 f32, K padded: d0,d1 | d2,0), pre-scaled by -2 so the
  // WMMA with C = |a|^2+|b|^2 yields squared distances directly.
  v2f a;
  {
    const int r = row0 + l16;
    if (!hi) { a.x = -2.0f * PA[r * 3 + 0]; a.y = -2.0f * PA[r * 3 + 1]; }
    else     { a.x = -2.0f * PA[r * 3 + 2]; a.y = 0.0f; }
  }
  float sqr[8];
#pragma unroll
  for (int r = 0; r < 8; ++r) sqr[r] = SA[row0 + r + (hi ? 8 : 0)];

  // online logsumexp state, base-2 domain
  float mx[8], sm[8];
#pragma unroll
  for (int r = 0; r < 8; ++r) { mx[r] = -3.0e38f; sm[r] = 0.0f; }

  for (int m0 = 0; m0 < kM; m0 += 16) {
    const int m = m0 + l16;
    v2f bv;
    if (!hi) { bv.x = sP[m * 3 + 0]; bv.y = sP[m * 3 + 1]; }
    else     { bv.x = sP[m * 3 + 2]; bv.y = 0.0f; }
    const float sqc = sSQ[m];
    const float yK  = sYK[m];

    v8f c;
#pragma unroll
    for (int r = 0; r < 8; ++r) c[r] = sqr[r] + sqc;
    // d2 tile = (-2*A) x B + (|a|^2 + |b|^2)
    c = __builtin_amdgcn_wmma_f32_16x16x4_f32(false, a, false, bv,
                                              (short)0, c, false, false);
#pragma unroll
    for (int r = 0; r < 8; ++r) {
      float d2   = c[r] > 0.0f ? c[r] : 0.0f;
      float dist = __builtin_amdgcn_sqrtf(d2);
      float t    = __builtin_fmaf(dist, -kK, yK);      // score in log2 units
      float nm   = fmaxf(mx[r], t);
      sm[r] = __builtin_fmaf(sm[r], __builtin_amdgcn_exp2f(mx[r] - nm),
                             __builtin_amdgcn_exp2f(t - nm));
      mx[r] = nm;
    }
  }

  // reduce the 16 column-lanes of each half (masks < 16 never cross halves)
#pragma unroll
  for (int r = 0; r < 8; ++r) {
#pragma unroll
    for (int msk = 1; msk < 16; msk <<= 1) {
      float omx = __shfl_xor(mx[r], msk, 32);
      float osm = __shfl_xor(sm[r], msk, 32);
      float nm  = fmaxf(mx[r], omx);
      sm[r] = __builtin_fmaf(sm[r], __builtin_amdgcn_exp2f(mx[r] - nm),
                             osm * __builtin_amdgcn_exp2f(omx - nm));
      mx[r] = nm;
    }
  }

  if (l16 == 0) {  // lane 0 -> rows row0+0..7, lane 16 -> rows row0+8..15
#pragma unroll
    for (int r = 0; r < 8; ++r) {
      const int row = row0 + r + (hi ? 8 : 0);
      float lse = kLn2 * (mx[r] + __builtin_amdgcn_logf(sm[r]));
      float xn  = kEps * (logw - lse);
      float df  = fabsf(xn - X[row]);
      atomicMax(maxdiff, __float_as_uint(df));  // nonneg float: bits monotone
      X[row] = xn;
    }
  }
}

// ------------------------- convergence check (1 thread) --------------------
__global__ void conv_check(unsigned* __restrict__ md_u,
                           unsigned* __restrict__ md_v,
                           unsigned* __restrict__ done) {
  if (*done == 0u) {
    float du = __uint_as_float(*md_u);
    float dv = __uint_as_float(*md_v);
    if (du < kTol && dv < kTol) *done = 1u;
  }
  *md_u = 0u;
  *md_v = 0u;
}

// ------------- final pass: sum( exp((u+v-dist)/eps) * dist ) per batch -----
__global__ __launch_bounds__(kThreads) void emd_final(
    const float* __restrict__ p1, const float* __restrict__ p2,
    const float* __restrict__ sq1, const float* __restrict__ sq2,
    const float* __restrict__ u, const float* __restrict__ v,
    float* __restrict__ partials) {
  __shared__ float sP[kM * 3];
  __shared__ float sSQ[kM];
  __shared__ float sVK[kM];
  __shared__ float wsum[kWavesPerBlock];

  const int b    = blockIdx.y;
  const int lane = threadIdx.x & 31;
  const int wave = threadIdx.x >> 5;
  const int row0 = blockIdx.x * kRowsPerBlock + wave * kRowsPerWave;

  const float* PA = p1  + (size_t)b * kN * 3;
  const float* PB = p2  + (size_t)b * kM * 3;
  const float* SA = sq1 + (size_t)b * kN;
  const float* SB = sq2 + (size_t)b * kM;
  const float* U  = u   + (size_t)b * kN;
  const float* V  = v   + (size_t)b * kM;

  for (int i = threadIdx.x; i < kM; i += kThreads) {
    sSQ[i] = SB[i];
    sVK[i] = V[i] * kK;
  }
  for (int i = threadIdx.x; i < kM * 3; i += kThreads) sP[i] = PB[i];
  __syncthreads();

  const bool hi  = lane >= 16;
  const int  l16 = lane & 15;

  v2f a;
  {
    const int r = row0 + l16;
    if (!hi) { a.x = -2.0f * PA[r * 3 + 0]; a.y = -2.0f * PA[r * 3 + 1]; }
    else     { a.x = -2.0f * PA[r * 3 + 2]; a.y = 0.0f; }
  }
  float sqr[8], uK[8];
#pragma unroll
  for (int r = 0; r < 8; ++r) {
    const int row = row0 + r + (hi ? 8 : 0);
    sqr[r] = SA[row];
    uK[r]  = U[row] * kK;
  }

  float acc[8];
#pragma unroll
  for (int r = 0; r < 8; ++r) acc[r] = 0.0f;

  for (int m0 = 0; m0 < kM; m0 += 16) {
    const int m = m0 + l16;
    v2f bv;
    if (!hi) { bv.x = sP[m * 3 + 0]; bv.y = sP[m * 3 + 1]; }
    else     { bv.x = sP[m * 3 + 2]; bv.y = 0.0f; }
    const float sqc = sSQ[m];
    const float vK  = sVK[m];

    v8f c;
#pragma unroll
    for (int r = 0; r < 8; ++r) c[r] = sqr[r] + sqc;
    c = __builtin_amdgcn_wmma_f32_16x16x4_f32(false, a, false, bv,
                                              (short)0, c, false, false);
#pragma unroll
    for (int r = 0; r < 8; ++r) {
      float d2   = c[r] > 0.0f ? c[r] : 0.0f;
      float dist = __builtin_amdgcn_sqrtf(d2);
      float t    = __builtin_fmaf(dist, -kK, uK[r] + vK);  // log2(P)
      acc[r] = __builtin_fmaf(__builtin_amdgcn_exp2f(t), dist, acc[r]);
    }
  }

  float s = ((acc[0] + acc[1]) + (acc[2] + acc[3])) +
            ((acc[4] + acc[5]) + (acc[6] + acc[7]));
#pragma unroll
  for (int msk = 1; msk < 32; msk <<= 1) s += __shfl_xor(s, msk, 32);
  if (lane == 0) wsum[wave] = s;
  __syncthreads();
  if (threadIdx.x == 0) {
    float t = 0.0f;
    for (int w = 0; w < kWavesPerBlock; ++w) t += wsum[w];
    partials[b * kBlocksPerBatch + blockIdx.x] = t;
  }
}

// ------------------ deterministic fixed-order final reduce -----------------
__global__ void emd_reduce(const float* __restrict__ partials,
                           float* __restrict__ out) {
  if (threadIdx.x == 0 && blockIdx.x == 0) {
    float tot = 0.0f;
    for (int b = 0; b < kB; ++b) {
      float s = 0.0f;
      for (int k = 0; k < kBlocksPerBatch; ++k)
        s += partials[b * kBlocksPerBatch + k];
      tot += s;
    }
    out[0] = tot / (float)kB;
  }
}

// ---------------------------------------------------------------------------
extern "C" void kernel_launch(void* const* d_in, const int* in_sizes, int n_in,
                              void* d_out, int out_size, void* d_ws,
                              size_t ws_size, hipStream_t stream) {
  (void)in_sizes; (void)n_in; (void)out_size; (void)ws_size;
  const float* p1 = (const float*)d_in[0];
  const float* p2 = (const float*)d_in[1];
  float* out = (float*)d_out;

  // workspace layout (floats): u | v | sq1 | sq2 | partials | flags(4 uints)
  const size_t BN = (size_t)kB * kN;
  float* ws       = (float*)d_ws;
  float* u        = ws;
  float* v        = ws + BN;
  float* sq1      = ws + 2 * BN;
  float* sq2      = ws + 3 * BN;
  float* partials = ws + 4 * BN;
  unsigned* flags = (unsigned*)(ws + 4 * BN + kB * kBlocksPerBatch);
  unsigned* md_u  = flags + 0;
  unsigned* md_v  = flags + 1;
  unsigned* done  = flags + 2;

  const float log_a = logf(1.0f / (float)kN + 1e-8f);
  const float log_b = logf(1.0f / (float)kM + 1e-8f);

  emd_init<<<(int)((BN + kThreads - 1) / kThreads), kThreads, 0, stream>>>(
      p1, p2, u, v, sq1, sq2, partials, flags);

  dim3 grid(kBlocksPerBatch, kB);
  for (int it = 0; it < kMaxIter; ++it) {
    // u <- eps*(log_a - LSE_m((v - dist)/eps))
    sinkhorn_update<<<grid, kThreads, 0, stream>>>(p1, p2, sq1, sq2, v, u,
                                                   md_u, done, log_a);
    // v <- eps*(log_b - LSE_n((u - dist)/eps))
    sinkhorn_update<<<grid, kThreads, 0, stream>>>(p2, p1, sq2, sq1, u, v,
                                                   md_v, done, log_b);
    conv_check<<<1, 1, 0, stream>>>(md_u, md_v, done);
  }

  emd_final<<<grid, kThreads, 0, stream>>>(p1, p2, sq1, sq2, u, v, partials);
  emd_reduce<<<1, 1, 0, stream>>>(partials, out);
}